// ProcessorBlock_343597383945
// MI455X (gfx1250) — compile-verified
//
#include <hip/hip_runtime.h>
#include <hip/hip_bf16.h>
#include <math.h>

#define D 128
#define NEG_SLOPE 0.2f
#define LN_EPS 1e-5f

typedef __attribute__((ext_vector_type(2))) float v2f;
typedef __attribute__((ext_vector_type(8))) float v8f;

__device__ __forceinline__ float waveReduceSum(float v) {
  #pragma unroll
  for (int off = 16; off > 0; off >>= 1) v += __shfl_xor(v, off, 32);
  return v;
}

__device__ __forceinline__ void atomicMaxFloat(float* addr, float value) {
  unsigned int* ua = (unsigned int*)addr;
  unsigned int old = *ua, assumed;
  do {
    assumed = old;
    if (__uint_as_float(assumed) >= value) break;
    old = atomicCAS(ua, assumed, __float_as_uint(value));
  } while (old != assumed);
}

// ---------------------------------------------------------------------------
// v = W_edge @ att_edge   (128x128 * 128 -> 128), one block
// ---------------------------------------------------------------------------
__global__ void k_vedge(const float* __restrict__ W_edge,
                        const float* __restrict__ att_edge,
                        float* __restrict__ v) {
  int t = threadIdx.x;  // 0..127
  float s = 0.f;
  #pragma unroll 4
  for (int d = 0; d < D; ++d) s += W_edge[t * D + d] * att_edge[d];
  v[t] = s;
}

// ---------------------------------------------------------------------------
// h = x @ W via V_WMMA_F32_16X16X4_F32.
// grid.x = ceil(N/16) row-stripes; 8 waves/block each own one 16-col tile.
// A (16x4 f32): lanes 0-15 hold M=lane, K={k,k+1}; lanes 16-31 hold K={k+2,k+3}.
// B (4x16 f32): VGPR0 = row k (lanes<16) / row k+2 (lanes>=16); VGPR1 = +1.
// C/D (16x16 f32): VGPR g -> M = g + (lane>=16 ? 8 : 0), N = lane&15.
// Tail handling is wave-uniform so the hot path has unconditional stores.
// ---------------------------------------------------------------------------
__global__ void k_gemm(const float* __restrict__ x, const float* __restrict__ W,
                       float* __restrict__ h, int N) {
  const int wave = threadIdx.x >> 5;
  const int lane = threadIdx.x & 31;
  const int row0 = blockIdx.x * 16;
  const int col0 = wave * 16;

  int mA = row0 + (lane & 15);
  if (mA >= N) mA = N - 1;              // clamp only matters for the tail tile
  const int kSel = (lane >> 4) << 1;    // 0 or 2
  const int nB = col0 + (lane & 15);

  const float* __restrict__ xr = x + (size_t)mA * D;
  v8f c = {};
  #pragma unroll 8
  for (int k = 0; k < D; k += 4) {
    v2f a, b;
    a.x = xr[k + kSel];
    a.y = xr[k + kSel + 1];
    b.x = W[(k + kSel) * D + nB];
    b.y = W[(k + kSel + 1) * D + nB];
    c = __builtin_amdgcn_wmma_f32_16x16x4_f32(false, a, false, b, (short)0, c,
                                              false, false);
  }

  const int rBase = row0 + ((lane >> 4) << 3);
  const int cIdx = col0 + (lane & 15);
  float* __restrict__ hp = h + (size_t)rBase * D + cIdx;
  if (row0 + 16 <= N) {                 // uniform: full tile, no EXEC churn
    #pragma unroll
    for (int g = 0; g < 8; ++g) hp[(size_t)g * D] = c[g];
  } else {                              // tail tile only
    #pragma unroll
    for (int g = 0; g < 8; ++g)
      if (rBase + g < N) hp[(size_t)g * D] = c[g];
  }
}

// ---------------------------------------------------------------------------
// a_src[i] = h[i].att_src ; a_dst[i] = h[i].att_dst   (one wave per node)
// ---------------------------------------------------------------------------
__global__ void k_rowdots(const float* __restrict__ h,
                          const float* __restrict__ att_src,
                          const float* __restrict__ att_dst,
                          float* __restrict__ a_src, float* __restrict__ a_dst,
                          int N) {
  const int wid = (int)((blockIdx.x * (size_t)blockDim.x + threadIdx.x) >> 5);
  const int lane = threadIdx.x & 31;
  if (wid >= N) return;
  float4 hv = ((const float4*)(h + (size_t)wid * D))[lane];
  float4 sv = ((const float4*)att_src)[lane];
  float4 dv = ((const float4*)att_dst)[lane];
  float s = hv.x * sv.x + hv.y * sv.y + hv.z * sv.z + hv.w * sv.w;
  float d = hv.x * dv.x + hv.y * dv.y + hv.z * dv.z + hv.w * dv.w;
  s = waveReduceSum(s);
  d = waveReduceSum(d);
  if (lane == 0) { a_src[wid] = s; a_dst[wid] = d; }
}

// ---------------------------------------------------------------------------
// e_dot[e] = edge_attr[e].v ; scatter sum/count by dst (one wave per edge)
// Streams the big 307 MB edge_attr array exactly once.
// ---------------------------------------------------------------------------
__global__ void k_edge_dot(const float* __restrict__ edge_attr,
                           const int* __restrict__ dst,
                           const float* __restrict__ v,
                           float* __restrict__ e_dot,
                           float* __restrict__ loop_sum,
                           float* __restrict__ cnt, int E) {
  const int wid = (int)((blockIdx.x * (size_t)blockDim.x + threadIdx.x) >> 5);
  const int lane = threadIdx.x & 31;
  if (wid >= E) return;
  float4 a = ((const float4*)(edge_attr + (size_t)wid * D))[lane];
  float4 vv = ((const float4*)v)[lane];
  float s = a.x * vv.x + a.y * vv.y + a.z * vv.z + a.w * vv.w;
  s = waveReduceSum(s);
  if (lane == 0) {
    e_dot[wid] = s;
    int dd = dst[wid];
    unsafeAtomicAdd(&loop_sum[dd], s);
    unsafeAtomicAdd(&cnt[dd], 1.0f);
  }
}

// ---------------------------------------------------------------------------
// Self-loop z; also (non-atomically) seeds zmax[i] = z_self[i].
// ---------------------------------------------------------------------------
__global__ void k_self(const float* __restrict__ a_src,
                       const float* __restrict__ a_dst,
                       const float* __restrict__ loop_sum,
                       const float* __restrict__ cnt,
                       float* __restrict__ z_self, float* __restrict__ zmax,
                       int N) {
  int i = blockIdx.x * blockDim.x + threadIdx.x;
  if (i >= N) return;
  float c = cnt[i];
  float ae = loop_sum[i] / fmaxf(c, 1.0f);
  float z = a_src[i] + a_dst[i] + ae;
  z = (z > 0.f) ? z : NEG_SLOPE * z;
  z_self[i] = z;
  zmax[i] = z;
}

// ---------------------------------------------------------------------------
// Per-edge z (leaky-relu), stored in place over e_dot; segment max into zmax.
// ---------------------------------------------------------------------------
__global__ void k_zedge(const int* __restrict__ src, const int* __restrict__ dst,
                        const float* __restrict__ a_src,
                        const float* __restrict__ a_dst,
                        float* __restrict__ zbuf /* in: e_dot, out: z */,
                        float* __restrict__ zmax, int E) {
  int e = blockIdx.x * blockDim.x + threadIdx.x;
  if (e >= E) return;
  int d = dst[e];
  float z = a_src[src[e]] + a_dst[d] + zbuf[e];
  z = (z > 0.f) ? z : NEG_SLOPE * z;
  zbuf[e] = z;
  atomicMaxFloat(&zmax[d], z);
}

// ---------------------------------------------------------------------------
// Fused softmax-numerator scatter: acc[dst] += exp(z - zmax[dst]) * h[src],
// denom[dst] += exp(...).  One wave per edge, float4 per lane.
// ---------------------------------------------------------------------------
__global__ void k_scatter(const int* __restrict__ src, const int* __restrict__ dst,
                          const float* __restrict__ zbuf,
                          const float* __restrict__ zmax,
                          const float* __restrict__ h,
                          float* __restrict__ denom, float* __restrict__ acc,
                          int E) {
  const int wid = (int)((blockIdx.x * (size_t)blockDim.x + threadIdx.x) >> 5);
  const int lane = threadIdx.x & 31;
  if (wid >= E) return;
  const int s = src[wid];
  const int d = dst[wid];
  const float ez = __expf(zbuf[wid] - zmax[d]);
  if (lane == 0) unsafeAtomicAdd(&denom[d], ez);
  float4 hv = ((const float4*)(h + (size_t)s * D))[lane];
  float* ap = acc + (size_t)d * D + lane * 4;
  unsafeAtomicAdd(ap + 0, ez * hv.x);
  unsafeAtomicAdd(ap + 1, ez * hv.y);
  unsafeAtomicAdd(ap + 2, ez * hv.z);
  unsafeAtomicAdd(ap + 3, ez * hv.w);
}

// ---------------------------------------------------------------------------
// Final: add self-loop term, normalize softmax, + bias, LayerNorm -> out.
// One wave per node.
// ---------------------------------------------------------------------------
__global__ void k_final(const float* __restrict__ acc,
                        const float* __restrict__ denom,
                        const float* __restrict__ z_self,
                        const float* __restrict__ zmax,
                        const float* __restrict__ h,
                        const float* __restrict__ bias,
                        const float* __restrict__ gamma,
                        const float* __restrict__ beta,
                        float* __restrict__ out, int N) {
  const int wid = (int)((blockIdx.x * (size_t)blockDim.x + threadIdx.x) >> 5);
  const int lane = threadIdx.x & 31;
  if (wid >= N) return;
  const float ezs = __expf(z_self[wid] - zmax[wid]);
  const float inv = 1.0f / (denom[wid] + ezs);

  float4 av = ((const float4*)(acc + (size_t)wid * D))[lane];
  float4 hv = ((const float4*)(h + (size_t)wid * D))[lane];
  float4 bv = ((const float4*)bias)[lane];
  float4 r;
  r.x = (av.x + ezs * hv.x) * inv + bv.x;
  r.y = (av.y + ezs * hv.y) * inv + bv.y;
  r.z = (av.z + ezs * hv.z) * inv + bv.z;
  r.w = (av.w + ezs * hv.w) * inv + bv.w;

  float mu = waveReduceSum(r.x + r.y + r.z + r.w) * (1.0f / D);
  float4 cdev = {r.x - mu, r.y - mu, r.z - mu, r.w - mu};
  float var = waveReduceSum(cdev.x * cdev.x + cdev.y * cdev.y +
                            cdev.z * cdev.z + cdev.w * cdev.w) * (1.0f / D);
  float rstd = rsqrtf(var + LN_EPS);

  float4 gv = ((const float4*)gamma)[lane];
  float4 be = ((const float4*)beta)[lane];
  float4 o;
  o.x = cdev.x * rstd * gv.x + be.x;
  o.y = cdev.y * rstd * gv.y + be.y;
  o.z = cdev.z * rstd * gv.z + be.z;
  o.w = cdev.w * rstd * gv.w + be.w;
  ((float4*)(out + (size_t)wid * D))[lane] = o;
}

// ---------------------------------------------------------------------------
extern "C" void kernel_launch(void* const* d_in, const int* in_sizes, int n_in,
                              void* d_out, int out_size, void* d_ws, size_t ws_size,
                              hipStream_t stream) {
  const float* x        = (const float*)d_in[0];
  const int*   eidx     = (const int*)d_in[1];
  const float* edge_attr= (const float*)d_in[2];
  const float* W        = (const float*)d_in[3];
  const float* att_src  = (const float*)d_in[4];
  const float* att_dst  = (const float*)d_in[5];
  const float* W_edge   = (const float*)d_in[6];
  const float* att_edge = (const float*)d_in[7];
  const float* bias     = (const float*)d_in[8];
  const float* gamma    = (const float*)d_in[9];
  const float* beta     = (const float*)d_in[10];

  const int N = in_sizes[0] / D;
  const int E = in_sizes[1] / 2;
  const int* src = eidx;
  const int* dst = eidx + E;

  float* ws = (float*)d_ws;
  size_t o = 0;
  float* h        = ws + o; o += (size_t)N * D;
  float* acc      = ws + o; o += (size_t)N * D;   // zeroed region starts here
  float* denom    = ws + o; o += N;
  float* cnt      = ws + o; o += N;
  float* loop_sum = ws + o; o += N;               // zeroed region ends here
  float* a_src    = ws + o; o += N;
  float* a_dst    = ws + o; o += N;
  float* zmax     = ws + o; o += N;
  float* z_self   = ws + o; o += N;
  float* e_dot    = ws + o; o += E;               // reused as z per edge
  float* v        = ws + o; o += D;

  // zero acc + denom + cnt + loop_sum (contiguous)
  hipMemsetAsync(acc, 0, ((size_t)N * D + 3 * (size_t)N) * sizeof(float), stream);

  k_vedge<<<1, D, 0, stream>>>(W_edge, att_edge, v);
  k_gemm<<<(N + 15) / 16, 256, 0, stream>>>(x, W, h, N);
  k_rowdots<<<(N + 7) / 8, 256, 0, stream>>>(h, att_src, att_dst, a_src, a_dst, N);
  k_edge_dot<<<(E + 7) / 8, 256, 0, stream>>>(edge_attr, dst, v, e_dot, loop_sum, cnt, E);
  k_self<<<(N + 255) / 256, 256, 0, stream>>>(a_src, a_dst, loop_sum, cnt, z_self, zmax, N);
  k_zedge<<<(E + 255) / 256, 256, 0, stream>>>(src, dst, a_src, a_dst, e_dot, zmax, E);
  k_scatter<<<(E + 7) / 8, 256, 0, stream>>>(src, dst, e_dot, zmax, h, denom, acc, E);
  k_final<<<(N + 7) / 8, 256, 0, stream>>>(acc, denom, z_self, zmax, h, bias, gamma, beta,
                                           (float*)d_out, N);
}